// DistiledRegionLoss_1494648619526
// MI455X (gfx1250) — compile-verified
//
#include <hip/hip_runtime.h>
#include <math.h>

// ---------------- problem constants (from reference) ----------------
#define NKP   9            // keypoints
#define NCLS  13           // classes
#define NB    16           // batch
#define NH    76
#define NW    76
#define SPW   (NH*NW)      // 5776 cells (divisible by 16: 361 tiles)
#define NTILE (SPW/16)     // 361
#define NT    64           // padded target count (50 real, 4 WMMA tiles of 16)
#define NL    21           // 2*NK+3 columns per target row
#define TH_C    80.0f
#define SHARP_C 2.0f
#define SIL_C   0.6f
#define OBJ_C   5.0f
#define NOOBJ_C 1.0f
#define IMW_C   640.0f
#define IMH_C   480.0f
#define INV_DEN 0.15651764274966565f   // 1/(e^2 - 1)
#define LOG2E   1.4426950408889634f
#define LN2     0.6931471805599453f

typedef __attribute__((ext_vector_type(2))) float v2f;
typedef __attribute__((ext_vector_type(8))) float v8f;

// -------- single-instruction TRANS32 math (co-issues with VALU/WMMA) --------
__device__ __forceinline__ float fast_exp(float x)  { return __builtin_amdgcn_exp2f(x * LOG2E); }
__device__ __forceinline__ float fast_exp2(float x) { return __builtin_amdgcn_exp2f(x); }
__device__ __forceinline__ float fast_log(float x)  { return __builtin_amdgcn_logf(x) * LN2; }
__device__ __forceinline__ float fast_rcp(float x)  { return __builtin_amdgcn_rcpf(x); }
__device__ __forceinline__ float fast_sqrt(float x) { return __builtin_amdgcn_sqrtf(x); }
__device__ __forceinline__ float sigmoidf_(float x) { return fast_rcp(1.0f + fast_exp(-x)); }

// ---------------- zero-init scatter targets ----------------
__global__ void init_k(float* __restrict__ coordMask, float* __restrict__ clsMask,
                       float* __restrict__ tcls) {
    int idx = blockIdx.x * blockDim.x + threadIdx.x;
    if (idx < NB * SPW) {
        coordMask[idx] = 0.0f;
        clsMask[idx]   = 0.0f;
        tcls[idx]      = 0.0f;
    }
}

// ---------------- GT preprocessing: B-matrix operands + prefix validity ----------------
// B column for keypoint k, target t:  [-2*gx, -2*gy, 1, gx^2+gy^2]  (scaled to pixels)
__global__ void prep_gt(const float* __restrict__ target,
                        float* __restrict__ bx, float* __restrict__ by,
                        float* __restrict__ b2, float* __restrict__ valid) {
    int b = blockIdx.x;      // batch
    int t = threadIdx.x;     // 0..63 (targets; >=50 are padding)
    const float* row = target + (size_t)b * 50 * NL;
    float v = 0.0f;
    if (t < 50) {
        bool ok = true;                      // cumprod(tgt[:, :, 1] != 0)
        for (int u = 0; u <= t; ++u) ok = ok && (row[u * NL + 1] != 0.0f);
        v = ok ? 1.0f : 0.0f;
    }
    valid[b * NT + t] = v;
    for (int k = 0; k < NKP; ++k) {
        float gx = 0.0f, gy = 0.0f;
        if (t < 50) {
            gx = row[t * NL + 1 + 2 * k] * IMW_C;
            gy = row[t * NL + 2 + 2 * k] * IMH_C;
        }
        size_t o = (size_t)(b * NKP + k) * NT + t;
        bx[o] = -2.0f * gx;
        by[o] = -2.0f * gy;
        b2[o] = gx * gx + gy * gy;
    }
}

// ---------------- pairwise corner-confidence via V_WMMA_F32_16X16X4_F32 ----------------
// One wave per (batch, 16-cell tile). A row (cell m): [ax, ay, ax^2+ay^2, 1].
// A layout (16x4 f32): lanes 0-15 hold K=0 (.x) / K=1 (.y); lanes 16-31 hold K=2 / K=3.
// B layout (4x16 f32): VGPR0 = rows K=0 (lanes 0-15) / K=2 (lanes 16-31); VGPR1 = K=1 / K=3.
// D = dist^2 per (cell M, target N); C/D layout: d[r] = cell r (lanes 0-15) or 8+r (lanes 16-31),
// N = lane & 15.  All gt loads are unconditional (every lane address valid) so the wave stays
// fully convergent -- no saveexec around the WMMAs, selects are plain v_cndmask.
__global__ __launch_bounds__(32) void conf_wmma(
    const float* __restrict__ out,
    const float* __restrict__ bx, const float* __restrict__ by,
    const float* __restrict__ b2v, const float* __restrict__ valid,
    float* __restrict__ confMask) {
    const int tile = blockIdx.x;        // 0..360
    const int b    = blockIdx.y;        // 0..15
    const int lane = threadIdx.x;       // 0..31
    const int lm   = lane & 15;
    const bool hi  = lane >= 16;
    const int cell = tile * 16 + lm;
    const int ci   = cell % NW;
    const int cj   = cell / NW;
    const float* ob = out + (size_t)b * 32 * SPW;

    // stage gt operands toward L2/L0 early (global_prefetch_b8)
    __builtin_prefetch(bx  + (size_t)b * NKP * NT, 0, 3);
    __builtin_prefetch(by  + (size_t)b * NKP * NT, 0, 3);
    __builtin_prefetch(b2v + (size_t)b * NKP * NT, 0, 3);

    // conf(dist) = (exp2(K0 - K1*dist) - 1) * INV_DEN, gated on dist < TH
    const float K0 = SHARP_C * LOG2E;           // exp(SHARP*(1 - d/TH)) in exp2 form
    const float K1 = SHARP_C * LOG2E / TH_C;

    float acc[4][8];
#pragma unroll
    for (int t4 = 0; t4 < 4; ++t4)
#pragma unroll
        for (int m = 0; m < 8; ++m) acc[t4][m] = 0.0f;

#pragma unroll
    for (int k = 0; k < NKP; ++k) {
        float vx = ob[(2 * k) * SPW + cell];
        float vy = ob[(2 * k + 1) * SPW + cell];
        if (k == 0) { vx = sigmoidf_(vx); vy = sigmoidf_(vy); }   // uniform branch
        float ax = (vx + (float)ci) * (IMW_C / (float)NW);
        float ay = (vy + (float)cj) * (IMH_C / (float)NH);
        v2f A;
        A.x = hi ? (ax * ax + ay * ay) : ax;   // K=2 : K=0
        A.y = hi ? 1.0f : ay;                  // K=3 : K=1
        const float* bxk = bx  + (size_t)(b * NKP + k) * NT;
        const float* byk = by  + (size_t)(b * NKP + k) * NT;
        const float* b2k = b2v + (size_t)(b * NKP + k) * NT;
#pragma unroll
        for (int t4 = 0; t4 < 4; ++t4) {
            int t = t4 * 16 + lm;
            // unconditional loads in every lane -> no exec-masked load regions
            float bxv = bxk[t];
            float byv = byk[t];
            float b2x = b2k[t];
            v2f B;
            B.x = hi ? 1.0f : bxv;             // K=2 : K=0
            B.y = hi ? b2x  : byv;             // K=3 : K=1
            v8f c = {};
            v8f d = __builtin_amdgcn_wmma_f32_16x16x4_f32(
                false, A, false, B, (short)0, c, false, false);
#pragma unroll
            for (int m = 0; m < 8; ++m) {
                float d2   = fmaxf(d[m], 0.0f);
                float dist = fast_sqrt(d2);                       // v_sqrt_f32
                float e    = fast_exp2(fmaf(dist, -K1, K0));      // v_exp_f32
                float cc   = (e - 1.0f) * INV_DEN;
                acc[t4][m] += (dist < TH_C) ? cc : 0.0f;          // v_cndmask
            }
        }
    }

    // mean over keypoints, validity mask, max over the 4 target tiles (per lane)
    float rm[8];
#pragma unroll
    for (int m = 0; m < 8; ++m) rm[m] = 0.0f;
#pragma unroll
    for (int t4 = 0; t4 < 4; ++t4) {
        float vm = valid[b * NT + t4 * 16 + lm];
#pragma unroll
        for (int m = 0; m < 8; ++m)
            rm[m] = fmaxf(rm[m], acc[t4][m] * (1.0f / 9.0f) * vm);
    }
    // max over the 16 targets within each half-wave (xor<16 stays in-half)
#pragma unroll
    for (int sh = 1; sh < 16; sh <<= 1)
#pragma unroll
        for (int m = 0; m < 8; ++m)
            rm[m] = fmaxf(rm[m], __shfl_xor(rm[m], sh, 32));

    if (lm == 0) {   // lane 0 -> cells M=0..7, lane 16 -> cells M=8..15
        int mbase = hi ? 8 : 0;
#pragma unroll
        for (int m = 0; m < 8; ++m) {
            float cmv = (rm[m] > SIL_C) ? 0.0f : NOOBJ_C;
            confMask[(size_t)b * SPW + tile * 16 + mbase + m] = cmv;
        }
    }
}

// ---------------- sequential scatter (order-preserving overwrite semantics) ----------------
__global__ void scatter_k(const float* __restrict__ target,
                          float* __restrict__ coordMask, float* __restrict__ confMask,
                          float* __restrict__ clsMask, float* __restrict__ tcls) {
    int b = threadIdx.x;
    if (b >= NB) return;
    const float* row = target + (size_t)b * 50 * NL;
    bool vflag = true;
    for (int t = 0; t < 50; ++t) {
        float cls = row[t * NL + 0];
        float x   = row[t * NL + 1];
        float y   = row[t * NL + 2];
        vflag = vflag && (x != 0.0f);
        int gi  = (int)(x * (float)NW);
        int gj  = (int)(y * (float)NH);
        int git = vflag ? gi : NW;                 // invalid -> OOB -> dropped
        if (git >= 0 && git < NW && gj >= 0 && gj < NH) {
            size_t o = (size_t)b * SPW + gj * NW + git;
            coordMask[o] = 1.0f;
            confMask[o]  = OBJ_C;
            clsMask[o]   = 1.0f;
            tcls[o]      = cls;
        }
    }
}

// ---------------- elementwise losses + deterministic block reduction ----------------
__global__ __launch_bounds__(256) void loss_k(
    const float* __restrict__ out, const float* __restrict__ dt,
    const float* __restrict__ coordMask, const float* __restrict__ confMask,
    const float* __restrict__ clsMask, const float* __restrict__ tcls,
    float* __restrict__ partials) {
    __shared__ float red[256];
    int idx = blockIdx.x * 256 + threadIdx.x;      // exactly NB*SPW threads
    int b = idx / SPW, s = idx % SPW;
    const float* ob = out + (size_t)b * 32 * SPW + s;
    const float* db = dt  + (size_t)b * 32 * SPW + s;

    float sum = 0.0f;
    // coord loss: 0.5 * cm * sum_{c<18} (coords - tcoords)^2, channels 0,1 sigmoided
    {
        float sq = 0.0f;
#pragma unroll
        for (int c = 0; c < 18; ++c) {
            float o = ob[c * SPW], d = db[c * SPW];
            if (c < 2) { o = sigmoidf_(o); d = sigmoidf_(d); }
            float df = o - d;
            sq += df * df;
        }
        sum += 0.5f * coordMask[idx] * sq;
    }
    // conf loss: 0.5 * mask * (sig(o18) - sig(d18))^2   (sqrt(mask)^2 == mask)
    {
        float o = sigmoidf_(ob[18 * SPW]), d = sigmoidf_(db[18 * SPW]);
        float df = o - d;
        sum += 0.5f * confMask[idx] * df * df;
    }
    // cls loss: cls_mask * CE(logits, tcls)  (log-softmax in exp2/log2 form)
    {
        float lg[NCLS];
        float mx = -3.4e38f;
#pragma unroll
        for (int c = 0; c < NCLS; ++c) { lg[c] = ob[(19 + c) * SPW]; mx = fmaxf(mx, lg[c]); }
        float se = 0.0f;
#pragma unroll
        for (int c = 0; c < NCLS; ++c) se += fast_exp2((lg[c] - mx) * LOG2E);
        float lse = mx + fast_log(se);
        int tc = (int)tcls[idx];
        tc = tc < 0 ? 0 : (tc > NCLS - 1 ? NCLS - 1 : tc);
        sum += clsMask[idx] * (lse - lg[tc]);
    }

    red[threadIdx.x] = sum;
    __syncthreads();
    for (int off = 128; off > 0; off >>= 1) {
        if (threadIdx.x < off) red[threadIdx.x] += red[threadIdx.x + off];
        __syncthreads();
    }
    if (threadIdx.x == 0) partials[blockIdx.x] = red[0];
}

__global__ void final_k(const float* __restrict__ partials, float* __restrict__ outp) {
    if (threadIdx.x == 0 && blockIdx.x == 0) {
        float s = 0.0f;
        for (int i = 0; i < (NB * SPW) / 256; ++i) s += partials[i];   // fixed order
        outp[0] = s;
    }
}

// ---------------- launch ----------------
extern "C" void kernel_launch(void* const* d_in, const int* in_sizes, int n_in,
                              void* d_out, int out_size, void* d_ws, size_t ws_size,
                              hipStream_t stream) {
    const float* output = (const float*)d_in[0];   // (16, 32, 76, 76)
    const float* dtgt   = (const float*)d_in[1];   // (16, 32, 76, 76)
    const float* target = (const float*)d_in[2];   // (16, 1050)
    float* outp = (float*)d_out;

    float* ws = (float*)d_ws;
    size_t o = 0;
    float* confMask  = ws + o; o += (size_t)NB * SPW;
    float* coordMask = ws + o; o += (size_t)NB * SPW;
    float* clsMask   = ws + o; o += (size_t)NB * SPW;
    float* tcls      = ws + o; o += (size_t)NB * SPW;
    float* bx        = ws + o; o += (size_t)NB * NKP * NT;
    float* by        = ws + o; o += (size_t)NB * NKP * NT;
    float* b2        = ws + o; o += (size_t)NB * NKP * NT;
    float* valid     = ws + o; o += (size_t)NB * NT;
    float* partials  = ws + o; o += (NB * SPW) / 256;

    hipLaunchKernelGGL(init_k, dim3((NB * SPW + 255) / 256), dim3(256), 0, stream,
                       coordMask, clsMask, tcls);
    hipLaunchKernelGGL(prep_gt, dim3(NB), dim3(NT), 0, stream,
                       target, bx, by, b2, valid);
    hipLaunchKernelGGL(conf_wmma, dim3(NTILE, NB), dim3(32), 0, stream,
                       output, bx, by, b2, valid, confMask);
    hipLaunchKernelGGL(scatter_k, dim3(1), dim3(NB), 0, stream,
                       target, coordMask, confMask, clsMask, tcls);
    hipLaunchKernelGGL(loss_k, dim3((NB * SPW) / 256), dim3(256), 0, stream,
                       output, dtgt, coordMask, confMask, clsMask, tcls, partials);
    hipLaunchKernelGGL(final_k, dim3(1), dim3(1), 0, stream, partials, outp);
}